// LRGCN_Model_8581344657592
// MI455X (gfx1250) — compile-verified
//
#include <hip/hip_runtime.h>
#include <cstdint>
#include <cstddef>

#define NNODES 50000
#define NEDGES 800000
#define NB 3

typedef __attribute__((ext_vector_type(2))) float v2f;
typedef __attribute__((ext_vector_type(8))) float v8f;

// ---------------------------------------------------------------- utilities
__global__ void zero_kernel(float* __restrict__ p, size_t n) {
  size_t i = (size_t)blockIdx.x * blockDim.x + threadIdx.x;
  if (i < n) p[i] = 0.f;
}

// count edges per (relation, dst) segment
__global__ void count_kernel(const int* __restrict__ ei, const int* __restrict__ et,
                             float* __restrict__ cnt) {
  int e = blockIdx.x * blockDim.x + threadIdx.x;
  if (e >= NEDGES) return;
  int dst = ei[NEDGES + e];
  atomicAdd(&cnt[(size_t)et[e] * NNODES + dst], 1.0f);
}

// scatter-add xin[src] into agg[rel][dst][*].  Feature dim is the fast lane
// index: each wave reads a contiguous row slice of xin and its atomics hit
// consecutive addresses in one agg row (cacheline-coherent in L2).
__global__ void scatter_kernel(const int* __restrict__ ei, const int* __restrict__ et,
                               const float* __restrict__ xin, float* __restrict__ agg,
                               int dims) {
  size_t i = (size_t)blockIdx.x * blockDim.x + threadIdx.x;
  size_t e = i >> 6;
  int d = (int)(i & 63);
  if (e >= NEDGES || d >= dims) return;
  int src = ei[e];
  int dst = ei[NEDGES + e];
  atomicAdd(&agg[((size_t)et[e] * NNODES + dst) * 64 + d],
            xin[(size_t)src * 64 + d]);
}

// agg /= max(cnt,1)
__global__ void normalize_kernel(float* __restrict__ agg, const float* __restrict__ cnt) {
  size_t i = (size_t)blockIdx.x * blockDim.x + threadIdx.x;
  if (i >= (size_t)4 * NNODES * 64) return;
  float c = cnt[i >> 6];
  agg[i] = agg[i] / fmaxf(c, 1.0f);
}

// ------------------------------------------------------- weight construction
struct GateP {
  const float* basis[3];  // [NB, cin, cg]
  const float* comp[3];   // [4, NB]
  const float* root[3];   // [cin, cg]
};
struct BiasP {
  const float* bx[3];
  const float* bh[3];
};

// Wcat[k][c], k in [0,320): chunks 0..3 = relation weights (W_r = sum_b comp*basis),
// chunk 4 = root. Columns: 3 gates of width cg, rest zero. Rows kk>=cin zero.
__global__ void buildW_kernel(GateP gp, float* __restrict__ Wcat, int cin, int cg, int npad) {
  int t = blockIdx.x * blockDim.x + threadIdx.x;
  int total = 320 * npad;
  if (t >= total) return;
  int k = t / npad, c = t - (t / npad) * npad;
  float val = 0.f;
  int g = c / cg;
  if (g < 3) {
    int cc = c - g * cg;
    int chunk = k >> 6, kk = k & 63;
    if (kk < cin) {
      if (chunk < 4) {
        float s = 0.f;
        for (int b = 0; b < NB; ++b)
          s += gp.comp[g][chunk * NB + b] * gp.basis[g][(b * cin + kk) * cg + cc];
        val = s;
      } else {
        val = gp.root[g][kk * cg + cc];
      }
    }
  }
  Wcat[(size_t)k * npad + c] = val;
}

// bsum[g*64+j] = bias_x[g][j] + bias_h[g][j]  (h-side conv == its bias since H=0)
__global__ void bias_kernel(BiasP bp, float* __restrict__ bsum, int cg) {
  int t = threadIdx.x;
  if (t >= 192) return;
  int g = t >> 6, j = t & 63;
  bsum[t] = (j < cg) ? (bp.bx[g][j] + bp.bh[g][j]) : 0.f;
}

// ------------------------------------------------------------- WMMA GEMM
// out[M, NT*16 cols at coff] = [mean chunks | xin] @ Wcat[:, coff:coff+NT*16]
// 8 waves/block, one 16-row M-tile per wave.  B panel staged per 64-row K-chunk
// in LDS, repacked as [k-group][col][4] so a lane's (k,k+1) pair is one
// aligned ds_load_b64.  A (global) and B (LDS) for group g+1 are loaded into
// fresh registers before the NT-deep WMMA chain for group g (double buffer).
template <int NT>
__global__ __launch_bounds__(256) void rgcn_gemm_wmma(const float* __restrict__ agg,
                                                      const float* __restrict__ xin,
                                                      const float* __restrict__ Wcat,
                                                      float* __restrict__ outBuf,
                                                      int wstride, int coff, int mtiles) {
  constexpr int NC = NT * 16;
  __shared__ float Bs[64 * NC];  // 16 k-groups x NC cols x 4 k-values

  const int wave = threadIdx.x >> 5;
  const int lane = threadIdx.x & 31;
  const int tile = blockIdx.x * 8 + wave;
  const bool active = (tile < mtiles);
  const int m0 = tile * 16;
  const int row = m0 + (lane & 15);   // A: M index
  const int kb = (lane >> 4) << 1;    // A/B: K sub-offset {0,2}
  const int col0 = lane & 15;         // B/C: N index within tile

  v8f acc[NT] = {};

#pragma unroll
  for (int chunk = 0; chunk < 5; ++chunk) {
    __syncthreads();
    for (int idx = threadIdx.x; idx < 64 * NC; idx += 256) {
      int k = idx / NC;
      int c = idx - k * NC;
      Bs[(k >> 2) * (NC * 4) + c * 4 + (k & 3)] =
          Wcat[(size_t)(chunk * 64 + k) * wstride + coff + c];
    }
    __syncthreads();

    if (active) {
      const float* abase = (chunk < 4)
                               ? agg + ((size_t)chunk * NNODES + row) * 64 + kb
                               : xin + (size_t)row * 64 + kb;
      const float* bs = Bs + col0 * 4 + kb;

      v2f a_cur = *(const v2f*)abase;
      v2f b_cur[NT];
#pragma unroll
      for (int t = 0; t < NT; ++t) b_cur[t] = *(const v2f*)(bs + (t * 16) * 4);

#pragma unroll
      for (int g = 0; g < 16; ++g) {
        v2f a_nxt = a_cur;
        v2f b_nxt[NT];
#pragma unroll
        for (int t = 0; t < NT; ++t) b_nxt[t] = b_cur[t];
        if (g < 15) {
          a_nxt = *(const v2f*)(abase + (g + 1) * 4);
#pragma unroll
          for (int t = 0; t < NT; ++t)
            b_nxt[t] = *(const v2f*)(bs + ((g + 1) * NC + t * 16) * 4);
        }
#pragma unroll
        for (int t = 0; t < NT; ++t)
          acc[t] = __builtin_amdgcn_wmma_f32_16x16x4_f32(
              false, a_cur, false, b_cur[t], (short)0, acc[t], false, false);
        a_cur = a_nxt;
#pragma unroll
        for (int t = 0; t < NT; ++t) b_cur[t] = b_nxt[t];
      }
    }
  }

  if (active) {
    const int rbase = m0 + ((lane >> 4) << 3);  // C: VGPR j -> row j (+8 hi lanes)
#pragma unroll
    for (int t = 0; t < NT; ++t) {
#pragma unroll
      for (int j = 0; j < 8; ++j) {
        outBuf[(size_t)(rbase + j) * wstride + coff + t * 16 + col0] = acc[t][j];
      }
    }
  }
}

// ---------------------------------------------------------- LSTM gate fusion
// H = sigmoid(O+bO) * tanh( sigmoid(I+bI) * tanh(T+bT) ); pad cols [cg,64) with 0
__global__ void gate_kernel(const float* __restrict__ gbuf, const float* __restrict__ bsum,
                            float* __restrict__ hout, int cg, int npad) {
  size_t i = (size_t)blockIdx.x * blockDim.x + threadIdx.x;
  if (i >= (size_t)NNODES * 64) return;
  int n = (int)(i >> 6), j = (int)(i & 63);
  float h = 0.f;
  if (j < cg) {
    const float* g = gbuf + (size_t)n * npad;
    float I = 1.0f / (1.0f + expf(-(g[j] + bsum[j])));
    float T = tanhf(g[cg + j] + bsum[64 + j]);
    float O = 1.0f / (1.0f + expf(-(g[2 * cg + j] + bsum[128 + j])));
    h = O * tanhf(I * T);
  }
  hout[i] = h;
}

// out[n] = relu(h2[n,0:20]) @ lw + lb
__global__ void final_kernel(const float* __restrict__ h2, const float* __restrict__ lw,
                             const float* __restrict__ lb, float* __restrict__ out) {
  int n = blockIdx.x * blockDim.x + threadIdx.x;
  if (n >= NNODES) return;
  const float* h = h2 + (size_t)n * 64;
  float acc = lb[0];
#pragma unroll
  for (int j = 0; j < 20; ++j) acc += fmaxf(h[j], 0.f) * lw[j];
  out[n] = acc;
}

// ------------------------------------------------------------------- driver
extern "C" void kernel_launch(void* const* d_in, const int* in_sizes, int n_in,
                              void* d_out, int out_size, void* d_ws, size_t ws_size,
                              hipStream_t stream) {
  (void)in_sizes; (void)n_in; (void)out_size; (void)ws_size;

  const float* x  = (const float*)d_in[0];
  const int*   ei = (const int*)d_in[1];
  const int*   et = (const int*)d_in[2];
  // p1 leaves start at 3, p2 at 35; each gate dict = {basis,comp,root,bias}
  // gate order: xi(0) hi(1) xf(2) hf(3) xc(4) hc(5) xo(6) ho(7)
  const int base1 = 3, base2 = 35;
  const float* lw = (const float*)d_in[67];
  const float* lb = (const float*)d_in[68];

  const int gx[3] = {0, 4, 6};  // xi, xc, xo -> I, T, O
  const int gh[3] = {1, 5, 7};  // hi, hc, ho (bias only)

  GateP gp1, gp2; BiasP bp1, bp2;
  for (int t = 0; t < 3; ++t) {
    gp1.basis[t] = (const float*)d_in[base1 + gx[t] * 4 + 0];
    gp1.comp[t]  = (const float*)d_in[base1 + gx[t] * 4 + 1];
    gp1.root[t]  = (const float*)d_in[base1 + gx[t] * 4 + 2];
    bp1.bx[t]    = (const float*)d_in[base1 + gx[t] * 4 + 3];
    bp1.bh[t]    = (const float*)d_in[base1 + gh[t] * 4 + 3];
    gp2.basis[t] = (const float*)d_in[base2 + gx[t] * 4 + 0];
    gp2.comp[t]  = (const float*)d_in[base2 + gx[t] * 4 + 1];
    gp2.root[t]  = (const float*)d_in[base2 + gx[t] * 4 + 2];
    bp2.bx[t]    = (const float*)d_in[base2 + gx[t] * 4 + 3];
    bp2.bh[t]    = (const float*)d_in[base2 + gh[t] * 4 + 3];
  }

  // workspace layout (floats)
  float* ws = (float*)d_ws;
  size_t off = 0;
  float* agg  = ws + off; off += (size_t)4 * NNODES * 64;   // 12.8M
  float* cnt  = ws + off; off += (size_t)4 * NNODES;        // 0.2M
  float* Wcat = ws + off; off += 320 * 160;
  float* bsum = ws + off; off += 256;
  float* gbuf = ws + off; off += (size_t)NNODES * 160;      // 8M
  float* h1   = ws + off; off += (size_t)NNODES * 64;       // 3.2M
  float* h2   = ws + off; off += (size_t)NNODES * 64;       // 3.2M

  const int B = 256;
  const size_t aggN = (size_t)4 * NNODES * 64;
  const size_t cntN = (size_t)4 * NNODES;
  const int mtiles = NNODES / 16;                 // 3125 (exact)
  const int gemmBlocks = (mtiles + 7) / 8;        // 391
  const int scatBlocks = (int)(((size_t)NEDGES * 64 + B - 1) / B);

  // ---- shared: edge counts
  zero_kernel<<<(int)((aggN + cntN + B - 1) / B), B, 0, stream>>>(agg, aggN + cntN);
  count_kernel<<<(NEDGES + B - 1) / B, B, 0, stream>>>(ei, et, cnt);

  // ---- layer 1  (cin=64, cg=50, npad=160)
  scatter_kernel<<<scatBlocks, B, 0, stream>>>(ei, et, x, agg, 64);
  normalize_kernel<<<(int)((aggN + B - 1) / B), B, 0, stream>>>(agg, cnt);
  buildW_kernel<<<(320 * 160 + B - 1) / B, B, 0, stream>>>(gp1, Wcat, 64, 50, 160);
  bias_kernel<<<1, 192, 0, stream>>>(bp1, bsum, 50);
  rgcn_gemm_wmma<5><<<gemmBlocks, 256, 0, stream>>>(agg, x, Wcat, gbuf, 160, 0, mtiles);
  rgcn_gemm_wmma<5><<<gemmBlocks, 256, 0, stream>>>(agg, x, Wcat, gbuf, 160, 80, mtiles);
  gate_kernel<<<(int)(((size_t)NNODES * 64 + B - 1) / B), B, 0, stream>>>(gbuf, bsum, h1, 50, 160);

  // ---- layer 2  (cin=50, cg=20, npad=64)
  zero_kernel<<<(int)((aggN + B - 1) / B), B, 0, stream>>>(agg, aggN);
  scatter_kernel<<<scatBlocks, B, 0, stream>>>(ei, et, h1, agg, 50);
  normalize_kernel<<<(int)((aggN + B - 1) / B), B, 0, stream>>>(agg, cnt);
  buildW_kernel<<<(320 * 64 + B - 1) / B, B, 0, stream>>>(gp2, Wcat, 50, 20, 64);
  bias_kernel<<<1, 192, 0, stream>>>(bp2, bsum, 20);
  rgcn_gemm_wmma<4><<<gemmBlocks, 256, 0, stream>>>(agg, h1, Wcat, gbuf, 64, 0, mtiles);
  gate_kernel<<<(int)(((size_t)NNODES * 64 + B - 1) / B), B, 0, stream>>>(gbuf, bsum, h2, 20, 64);

  // ---- relu + linear head
  final_kernel<<<(NNODES + B - 1) / B, B, 0, stream>>>(h2, lw, lb, (float*)d_out);
}